// GAT_79585743995154
// MI455X (gfx1250) — compile-verified
//
#include <hip/hip_runtime.h>
#include <hip/hip_bf16.h>

typedef __attribute__((ext_vector_type(2))) float v2f;
typedef __attribute__((ext_vector_type(8))) float v8f;

#define N_NODES 20000
#define N_EDGES 640000
#define DIMH    128

// ---------------------------------------------------------------------------
// Fold W_dst @ att_dst and W_src @ att_src into 128-vectors (one tiny block).
// ---------------------------------------------------------------------------
__global__ void gat_att_fold(const float* __restrict__ W_src,
                             const float* __restrict__ W_dst,
                             const float* __restrict__ att_src,
                             const float* __restrict__ att_dst,
                             float* __restrict__ vs, float* __restrict__ vd) {
    int d = threadIdx.x; // 128 threads
    float s = 0.f, t = 0.f;
    for (int h = 0; h < DIMH; ++h) {
        s += W_src[d * DIMH + h] * att_src[h];
        t += W_dst[d * DIMH + h] * att_dst[h];
    }
    vs[d] = s;
    vd[d] = t;
}

// ---------------------------------------------------------------------------
// Init: zero agg [N,128], zero denom, seg_max keys = 0 (== -inf in key space)
// ---------------------------------------------------------------------------
__global__ void gat_init(unsigned* __restrict__ segkey,
                         float* __restrict__ denom,
                         float* __restrict__ agg) {
    int i = blockIdx.x * blockDim.x + threadIdx.x;
    if (i < N_NODES * DIMH) agg[i] = 0.f;
    if (i < N_NODES) { segkey[i] = 0u; denom[i] = 0.f; }
}

// ---------------------------------------------------------------------------
// FP32 WMMA GEMM:  C[M,128] = f(A[M,128]) @ B[128,128] (+ out_bias)
//   FUSE=1: f(a) = relu(a + in_bias[k]), and C += out_bias[col]
// Block covers 32 rows x 128 cols. Wave w owns cols [16w,16w+16) and TWO
// 16x16 accumulators (rows row0..+16 and row0+16..+32): two independent
// V_WMMA_F32_16X16X4_F32 chains sharing each B fragment. Branch-free body.
// ---------------------------------------------------------------------------
template <bool FUSE>
__global__ __launch_bounds__(256)
void gat_gemm128(const float* __restrict__ A, const float* __restrict__ B,
                 float* __restrict__ C,
                 const float* __restrict__ in_bias,
                 const float* __restrict__ out_bias) {
    const int wave  = threadIdx.x >> 5;
    const int lane  = threadIdx.x & 31;
    const int hi    = lane >> 4;          // 0: lanes 0-15, 1: lanes 16-31
    const int lo    = lane & 15;
    const int row0  = blockIdx.x * 32;
    const int col   = (wave << 4) + lo;   // output column (0..127)
    const int arow0 = row0 + lo;          // A row for tile 0
    const int arow1 = row0 + 16 + lo;     // A row for tile 1

    v8f acc0 = {}, acc1 = {};
#pragma unroll
    for (int k0 = 0; k0 < DIMH; k0 += 4) {
        const int ka = k0 + 2 * hi;       // this lane's K pair
        // A fragments (16x4 each): lane holds {A[row][ka], A[row][ka+1]}
        v2f a0 = *(const v2f*)(A + arow0 * DIMH + ka);
        v2f a1 = *(const v2f*)(A + arow1 * DIMH + ka);
        if (FUSE) {
            const float b0 = in_bias[ka];
            const float b1 = in_bias[ka + 1];
            a0.x = fmaxf(a0.x + b0, 0.f);  a0.y = fmaxf(a0.y + b1, 0.f);
            a1.x = fmaxf(a1.x + b0, 0.f);  a1.y = fmaxf(a1.y + b1, 0.f);
        }
        // B fragment (4x16): lane holds {B[ka][col], B[ka+1][col]} (shared)
        v2f b;
        b.x = B[ka * DIMH + col];
        b.y = B[(ka + 1) * DIMH + col];
        acc0 = __builtin_amdgcn_wmma_f32_16x16x4_f32(
            false, a0, false, b, (short)0, acc0, false, false);
        acc1 = __builtin_amdgcn_wmma_f32_16x16x4_f32(
            false, a1, false, b, (short)0, acc1, false, false);
    }
    const float ob = FUSE ? out_bias[col] : 0.f;
#pragma unroll
    for (int r = 0; r < 8; ++r) {
        // D layout: VGPR r -> row r + 8*hi within the 16x16 tile
        C[(row0 + r + 8 * hi) * DIMH + col]      = acc0[r] + ob;
        C[(row0 + 16 + r + 8 * hi) * DIMH + col] = acc1[r] + ob;
    }
}

// ---------------------------------------------------------------------------
// Per-node attention scalars: a_src[n] = x[n,:].vs ; a_dst[n] = x[n,:].vd
// One wave32 per node, float4 per lane, shuffle reduction.
// ---------------------------------------------------------------------------
__global__ __launch_bounds__(256)
void gat_node_att(const float* __restrict__ x,
                  const float* __restrict__ vs, const float* __restrict__ vd,
                  float* __restrict__ a_src, float* __restrict__ a_dst) {
    const int wave = threadIdx.x >> 5;
    const int lane = threadIdx.x & 31;
    const int n = blockIdx.x * 8 + wave;
    if (n >= N_NODES) return;
    float4 xv = ((const float4*)(x + (size_t)n * DIMH))[lane];
    float4 s4 = ((const float4*)vs)[lane];
    float4 d4 = ((const float4*)vd)[lane];
    float s = xv.x * s4.x + xv.y * s4.y + xv.z * s4.z + xv.w * s4.w;
    float d = xv.x * d4.x + xv.y * d4.y + xv.z * d4.z + xv.w * d4.w;
    for (int off = 16; off > 0; off >>= 1) {
        s += __shfl_xor(s, off, 32);
        d += __shfl_xor(d, off, 32);
    }
    if (lane == 0) { a_src[n] = s; a_dst[n] = d; }
}

// --- order-preserving float <-> uint key mapping (for atomic max) ----------
__device__ __forceinline__ unsigned f2key(float f) {
    unsigned b = __float_as_uint(f);
    return (b & 0x80000000u) ? ~b : (b | 0x80000000u);
}
__device__ __forceinline__ float key2f(unsigned k) {
    unsigned b = (k & 0x80000000u) ? (k ^ 0x80000000u) : ~k;
    return __uint_as_float(b);
}

// ---------------------------------------------------------------------------
// Edge pass 1: logits + segment max (uint-key atomicMax in L2)
// ---------------------------------------------------------------------------
__global__ __launch_bounds__(256)
void gat_edge_logits(const int* __restrict__ src, const int* __restrict__ dst,
                     const float* __restrict__ a_src,
                     const float* __restrict__ a_dst,
                     float* __restrict__ logits, unsigned* __restrict__ segkey) {
    int e = blockIdx.x * blockDim.x + threadIdx.x;
    if (e >= N_EDGES) return;
    int s = src[e], d = dst[e];
    float l = a_src[s] + a_dst[d];
    l = l > 0.f ? l : 0.2f * l;          // leaky_relu(0.2)
    logits[e] = l;
    atomicMax(&segkey[d], f2key(l));
}

// ---------------------------------------------------------------------------
// Edge pass 2: e = exp(logit - max[dst]); denom[dst] += e
// ---------------------------------------------------------------------------
__global__ __launch_bounds__(256)
void gat_edge_exp(const int* __restrict__ dst,
                  const float* __restrict__ logits,
                  const unsigned* __restrict__ segkey,
                  float* __restrict__ evals, float* __restrict__ denom) {
    int e = blockIdx.x * blockDim.x + threadIdx.x;
    if (e >= N_EDGES) return;
    int d = dst[e];
    float ev = __expf(logits[e] - key2f(segkey[d]));
    evals[e] = ev;
    unsafeAtomicAdd(&denom[d], ev);
}

// ---------------------------------------------------------------------------
// Edge pass 3: agg[dst,:] += alpha * h_src[src,:]
// One wave32 per edge; lane handles 4 contiguous floats (float4 gather from
// L2-resident h_src, 4 hardware fp32 atomic adds into L2-resident agg).
// ---------------------------------------------------------------------------
__global__ __launch_bounds__(256)
void gat_edge_scatter(const int* __restrict__ src, const int* __restrict__ dst,
                      const float* __restrict__ evals,
                      const float* __restrict__ denom,
                      const float* __restrict__ h_src,
                      float* __restrict__ agg) {
    const int wave = threadIdx.x >> 5;
    const int lane = threadIdx.x & 31;
    const int e = blockIdx.x * 8 + wave;
    if (e >= N_EDGES) return;
    int s = src[e], d = dst[e];
    float alpha = evals[e] / denom[d];
    float4 h = ((const float4*)(h_src + (size_t)s * DIMH))[lane];
    float* ap = agg + (size_t)d * DIMH + lane * 4;
    unsafeAtomicAdd(ap + 0, alpha * h.x);
    unsafeAtomicAdd(ap + 1, alpha * h.y);
    unsafeAtomicAdd(ap + 2, alpha * h.z);
    unsafeAtomicAdd(ap + 3, alpha * h.w);
}

// ---------------------------------------------------------------------------
extern "C" void kernel_launch(void* const* d_in, const int* in_sizes, int n_in,
                              void* d_out, int out_size, void* d_ws, size_t ws_size,
                              hipStream_t stream) {
    const float* x       = (const float*)d_in[0];
    const int*   eidx    = (const int*)d_in[1];   // [2, E] int32
    const float* W_src   = (const float*)d_in[2];
    const float* W_dst   = (const float*)d_in[3];
    const float* att_src = (const float*)d_in[4];
    const float* att_dst = (const float*)d_in[5];
    const float* bias    = (const float*)d_in[6];
    const float* W_lin   = (const float*)d_in[7];
    const float* b_lin   = (const float*)d_in[8];
    float* out = (float*)d_out;

    const int* src = eidx;
    const int* dst = eidx + N_EDGES;

    // workspace layout (bytes)
    char* base = (char*)d_ws;
    float*    h_src_w = (float*)(base + 0);                  // 10,240,000
    float*    agg     = (float*)(base + 10240000);           // 10,240,000
    float*    logits  = (float*)(base + 20480000);           //  2,560,000
    float*    evals   = (float*)(base + 23040000);           //  2,560,000
    float*    a_src   = (float*)(base + 25600000);           //     80,000
    float*    a_dst   = (float*)(base + 25680000);           //     80,000
    unsigned* segkey  = (unsigned*)(base + 25760000);        //     80,000
    float*    denom   = (float*)(base + 25840000);           //     80,000
    float*    vs      = (float*)(base + 25920000);           //        512
    float*    vd      = (float*)(base + 25920512);           //        512

    // fold attention vectors through the projection weights
    gat_att_fold<<<1, 128, 0, stream>>>(W_src, W_dst, att_src, att_dst, vs, vd);

    // init accumulators
    gat_init<<<(N_NODES * DIMH + 255) / 256, 256, 0, stream>>>(segkey, denom, agg);

    // h_src = x @ W_src   (fp32 WMMA, no epilogue)
    gat_gemm128<false><<<N_NODES / 32, 256, 0, stream>>>(x, W_src, h_src_w,
                                                         nullptr, nullptr);

    // per-node attention scalars
    gat_node_att<<<N_NODES / 8, 256, 0, stream>>>(x, vs, vd, a_src, a_dst);

    // edge phase
    gat_edge_logits<<<N_EDGES / 256, 256, 0, stream>>>(src, dst, a_src, a_dst, logits, segkey);
    gat_edge_exp<<<N_EDGES / 256, 256, 0, stream>>>(dst, logits, segkey, evals, denom);
    gat_edge_scatter<<<N_EDGES / 8, 256, 0, stream>>>(src, dst, evals, denom, h_src_w, agg);

    // out = relu(agg + bias) @ W_lin + b_lin   (fp32 WMMA, fused epilogue)
    gat_gemm128<true><<<N_NODES / 32, 256, 0, stream>>>(agg, W_lin, out,
                                                        bias, b_lin);
}